// ConvAutoencoderGroupsHyp_87754771792380
// MI455X (gfx1250) — compile-verified
//
#include <hip/hip_runtime.h>
#include <hip/hip_bf16.h>

typedef __attribute__((ext_vector_type(2)))  float v2f;
typedef __attribute__((ext_vector_type(8)))  float v8f;

#define TPB  256
#define TILE 16
#define TDIM 18   // 16 + 2 halo

// ---- CDNA5 async global->LDS path (guarded; falls back to plain ds stores) --
#if defined(__has_builtin)
#if __has_builtin(__builtin_amdgcn_global_load_async_to_lds_b32)
#define HAVE_ASYNC_LDS 1
#endif
#endif
#ifndef HAVE_ASYNC_LDS
#define HAVE_ASYNC_LDS 0
#endif

#if HAVE_ASYNC_LDS
typedef __attribute__((address_space(1))) int* GPTR;   // global src (int*)
typedef __attribute__((address_space(3))) int* LPTR;   // LDS dst
#endif

__device__ __forceinline__ void async_wait0()
{
#if HAVE_ASYNC_LDS
#if __has_builtin(__builtin_amdgcn_s_wait_asynccnt)
    __builtin_amdgcn_s_wait_asynccnt(0);
#else
    asm volatile("s_wait_asynccnt 0x0" ::: "memory");
#endif
#endif
}

// ---------------------------------------------------------------------------
// Grouped 3x3 conv (pad=1) + bias with LDS input tiling.
// grid: (#16x16 tiles, G, N); block = 256 threads = one 16x16 output tile.
// The CinG input-channel halo tiles are staged into LDS once (async copy on
// CDNA5), then all CoutG output channels of the group are computed from LDS:
// global input traffic drops by 9*CoutG vs the direct form.
// Per-channel batch-norm sum/sumsq are block-reduced and atomically merged.
// ---------------------------------------------------------------------------
template <int CING, int COUTG>
__global__ void gconv3x3_tiled(const float* __restrict__ x,
                               const float* __restrict__ w,
                               const float* __restrict__ bias,
                               float* __restrict__ y,
                               float* __restrict__ s1, float* __restrict__ s2,
                               int H, int W, int G)
{
    __shared__ float tile[CING][TDIM * TDIM];
    __shared__ float red0[TPB];
    __shared__ float red1[TPB];

    const int g   = blockIdx.y;
    const int n   = blockIdx.z;
    const int tW  = (W + TILE - 1) / TILE;
    const int ty0 = (blockIdx.x / tW) * TILE;
    const int tx0 = (blockIdx.x % tW) * TILE;
    const int tid = threadIdx.x;
    const int ty  = tid >> 4, tx = tid & 15;
    const int oh  = ty0 + ty, ow = tx0 + tx;
    const bool valid = (oh < H) && (ow < W);

    // ---- stage input halo tiles into LDS ----
    #pragma unroll
    for (int ic = 0; ic < CING; ++ic) {
        const float* xp = x + (((size_t)n * (G * CING) + (size_t)g * CING + ic) * H) * W;
        for (int j = tid; j < TDIM * TDIM; j += TPB) {
            const int r  = j / TDIM, c = j - r * TDIM;
            const int ih = ty0 - 1 + r, iw = tx0 - 1 + c;
            const bool inb = (ih >= 0) && (ih < H) && (iw >= 0) && (iw < W);
#if HAVE_ASYNC_LDS
            if (!inb) tile[ic][j] = 0.0f;                 // halo: ds store
            if (inb)                                      // interior: async copy
                __builtin_amdgcn_global_load_async_to_lds_b32(
                    (GPTR)(xp + ih * W + iw), (LPTR)&tile[ic][j], 0, 0);
#else
            tile[ic][j] = inb ? xp[ih * W + iw] : 0.0f;
#endif
        }
    }
    async_wait0();
    __syncthreads();

    // ---- compute all CoutG outputs of this group from LDS ----
    float accs[COUTG];
    #pragma unroll
    for (int ocg = 0; ocg < COUTG; ++ocg) {
        const int oc = g * COUTG + ocg;
        const float* wp = w + (size_t)oc * CING * 9;
        float acc = bias[oc];
        #pragma unroll
        for (int ic = 0; ic < CING; ++ic) {
            const float* t = &tile[ic][ty * TDIM + tx];
            #pragma unroll
            for (int kh = 0; kh < 3; ++kh)
                #pragma unroll
                for (int kw = 0; kw < 3; ++kw)
                    acc = fmaf(t[kh * TDIM + kw], wp[ic * 9 + kh * 3 + kw], acc);
        }
        accs[ocg] = acc;
        if (valid)
            y[(((size_t)n * (G * COUTG) + oc) * H) * W + oh * W + ow] = acc;
    }

    // ---- per-output-channel block reduction for BN statistics ----
    #pragma unroll
    for (int ocg = 0; ocg < COUTG; ++ocg) {
        const float v = valid ? accs[ocg] : 0.0f;
        __syncthreads();
        red0[tid] = v;
        red1[tid] = v * v;
        __syncthreads();
        for (int s = TPB / 2; s > 0; s >>= 1) {
            if (tid < s) { red0[tid] += red0[tid + s]; red1[tid] += red1[tid + s]; }
            __syncthreads();
        }
        if (tid == 0) {
            atomicAdd(&s1[g * COUTG + ocg], red0[0]);
            atomicAdd(&s2[g * COUTG + ocg], red1[0]);
        }
    }
}

// ---------------------------------------------------------------------------
// Grouped ConvTranspose2d k=2 s=2 (non-overlapping) + bias + fused stats.
// torch weight layout: w[Cin, CoutG, 2, 2]. out[oh,ow] <- in[oh/2, ow/2].
// grid: (ceil(HoWo/TPB), Cout, N)
// ---------------------------------------------------------------------------
__global__ void gconvT2x2_stats(const float* __restrict__ x,
                                const float* __restrict__ w,
                                const float* __restrict__ bias,
                                float* __restrict__ y,
                                float* __restrict__ s1, float* __restrict__ s2,
                                int Hin, int Win, int CinG, int CoutG, int G)
{
    const int oc  = blockIdx.y;
    const int n   = blockIdx.z;
    const int g   = oc / CoutG;
    const int ocg = oc - g * CoutG;
    const int Ho = Hin * 2, Wo = Win * 2;
    const int HW = Ho * Wo;
    const int hw = blockIdx.x * TPB + threadIdx.x;

    float v = 0.0f;
    if (hw < HW) {
        const int oh = hw / Wo, ow = hw - oh * Wo;
        const int ih = oh >> 1, kh = oh & 1;
        const int iw = ow >> 1, kw = ow & 1;
        float acc = bias[oc];
        for (int ic = 0; ic < CinG; ++ic) {
            const float xv = x[(((size_t)n * (G * CinG) + (size_t)g * CinG + ic) * Hin + ih) * Win + iw];
            const float wv = w[((((size_t)g * CinG + ic) * CoutG + ocg) * 2 + kh) * 2 + kw];
            acc = fmaf(xv, wv, acc);
        }
        y[(((size_t)n * (G * CoutG) + oc) * Ho) * Wo + hw] = acc;
        v = acc;
    }

    __shared__ float ls[TPB];
    __shared__ float lq[TPB];
    ls[threadIdx.x] = v;
    lq[threadIdx.x] = v * v;
    __syncthreads();
    for (int s = TPB / 2; s > 0; s >>= 1) {
        if (threadIdx.x < s) {
            ls[threadIdx.x] += ls[threadIdx.x + s];
            lq[threadIdx.x] += lq[threadIdx.x + s];
        }
        __syncthreads();
    }
    if (threadIdx.x == 0) {
        atomicAdd(&s1[oc], ls[0]);
        atomicAdd(&s2[oc], lq[0]);
    }
}

// mean / invstd from accumulated sum / sumsq
__global__ void finalize_stats(const float* __restrict__ s1, const float* __restrict__ s2,
                               float* __restrict__ mean, float* __restrict__ istd,
                               int C, float invCount)
{
    const int c = blockIdx.x * TPB + threadIdx.x;
    if (c >= C) return;
    const float m = s1[c] * invCount;
    const float var = s2[c] * invCount - m * m;
    mean[c] = m;
    istd[c] = rsqrtf(var + 1e-5f);
}

// BN(train) + ReLU + 2x2 maxpool, one thread per pooled output pixel
__global__ void bn_relu_pool2(const float* __restrict__ y,
                              const float* __restrict__ mean, const float* __restrict__ istd,
                              const float* __restrict__ gamma, const float* __restrict__ beta,
                              float* __restrict__ out, int C, int H, int W)
{
    const int c = blockIdx.y;
    const int n = blockIdx.z;
    const int Ho = H >> 1, Wo = W >> 1;
    const int hw = blockIdx.x * TPB + threadIdx.x;
    if (hw >= Ho * Wo) return;
    const int oh = hw / Wo, ow = hw - oh * Wo;
    const float* p = y + (((size_t)n * C + c) * H + 2 * oh) * W + 2 * ow;
    const float m  = mean[c];
    const float sg = istd[c] * gamma[c];
    const float bt = beta[c];
    const float a = fmaxf(fmaf(p[0]     - m, sg, bt), 0.0f);
    const float b = fmaxf(fmaf(p[1]     - m, sg, bt), 0.0f);
    const float d = fmaxf(fmaf(p[W]     - m, sg, bt), 0.0f);
    const float e = fmaxf(fmaf(p[W + 1] - m, sg, bt), 0.0f);
    out[(((size_t)n * C + c) * Ho + oh) * Wo + ow] = fmaxf(fmaxf(a, b), fmaxf(d, e));
}

// BN + activation (0 = ReLU, 1 = sigmoid), flat elementwise
__global__ void bn_act(const float* __restrict__ y,
                       const float* __restrict__ mean, const float* __restrict__ istd,
                       const float* __restrict__ gamma, const float* __restrict__ beta,
                       float* __restrict__ out, size_t total, int C, int HW, int act)
{
    const size_t i = (size_t)blockIdx.x * TPB + threadIdx.x;
    if (i >= total) return;
    const int c = (int)((i / (size_t)HW) % (size_t)C);
    float v = fmaf(y[i] - mean[c], istd[c] * gamma[c], beta[c]);
    v = act ? (1.0f / (1.0f + __expf(-v))) : fmaxf(v, 0.0f);
    out[i] = v;
}

// BN + ReLU + 8x8 maxpool of [N,C,8,8] -> [N*C] (one thread per (n,c))
__global__ void bn_relu_pool8(const float* __restrict__ y,
                              const float* __restrict__ mean, const float* __restrict__ istd,
                              const float* __restrict__ gamma, const float* __restrict__ beta,
                              float* __restrict__ out, int NC, int C)
{
    const int i = blockIdx.x * TPB + threadIdx.x;
    if (i >= NC) return;
    const int c = i % C;
    const float* p = y + (size_t)i * 64;
    const float m  = mean[c];
    const float sg = istd[c] * gamma[c];
    const float bt = beta[c];
    float best = 0.0f;  // ReLU outputs are >= 0
    #pragma unroll
    for (int j = 0; j < 64; ++j)
        best = fmaxf(best, fmaxf(fmaf(p[j] - m, sg, bt), 0.0f));
    out[i] = best;
}

// ---------------------------------------------------------------------------
// Classifier: pred = sigmoid(flat[16,373] @ cw^T + cb) on the f32 matrix core
// (V_WMMA_F32_16X16X4_F32). M = 16 batch rows, weight column lives at N = 0.
// Branchless: 93 unguarded K=4 steps + one value-masked tail step, so no
// EXEC save/restore churn around the WMMA. One full wave (EXEC all ones).
// ---------------------------------------------------------------------------
__global__ void classifier_wmma(const float* __restrict__ flat,
                                const float* __restrict__ cw,
                                const float* __restrict__ cb,
                                float* __restrict__ pred, int K)
{
    const int lane = threadIdx.x;
    const int l    = lane & 15;    // M (for A) / N (for B,D)
    const int half = lane >> 4;
    const int Kmain = K & ~3;      // 372: all loads below provably in-range
    v8f acc = {};
    for (int k0 = 0; k0 < Kmain; k0 += 4) {
        const int ka = k0 + 2 * half;  // A 16x4 f32: VGPR0=K(2h), VGPR1=K(2h+1)
        v2f a, b;
        a.x = flat[l * K + ka];
        a.y = flat[l * K + ka + 1];
        const float w0 = cw[ka];       // lane-uniform loads, value-masked below
        const float w1 = cw[ka + 1];
        b.x = (l == 0) ? w0 : 0.0f;
        b.y = (l == 0) ? w1 : 0.0f;
        acc = __builtin_amdgcn_wmma_f32_16x16x4_f32(
            false, a, false, b, (short)0, acc, false, false);
    }
    {   // tail: single remaining k = K-1 (valid only in half 0, slot 0)
        const float av = flat[l * K + (K - 1)];
        const float wv = cw[K - 1];
        v2f a, b;
        a.x = (half == 0) ? av : 0.0f;
        a.y = 0.0f;
        b.x = (l == 0 && half == 0) ? wv : 0.0f;
        b.y = 0.0f;
        acc = __builtin_amdgcn_wmma_f32_16x16x4_f32(
            false, a, false, b, (short)0, acc, false, false);
    }
    // D layout: VGPR j, lane l -> (M = j + 8*half, N = l); take column N=0
    if (l == 0) {
        const float bias = cb[0];
        #pragma unroll
        for (int j = 0; j < 8; ++j) {
            const float v = acc[j] + bias;
            pred[j + 8 * half] = 1.0f / (1.0f + __expf(-v));
        }
    }
}

// ---------------------------------------------------------------------------
extern "C" void kernel_launch(void* const* d_in, const int* in_sizes, int n_in,
                              void* d_out, int out_size, void* d_ws, size_t ws_size,
                              hipStream_t stream)
{
    const int N = 16, G = 373;

    const float* x   = (const float*)d_in[0];
    const float* w1  = (const float*)d_in[1];
    const float* b1  = (const float*)d_in[2];
    const float* g1  = (const float*)d_in[3];
    const float* be1 = (const float*)d_in[4];
    const float* w2  = (const float*)d_in[5];
    const float* b2  = (const float*)d_in[6];
    const float* g2  = (const float*)d_in[7];
    const float* be2 = (const float*)d_in[8];
    const float* w3  = (const float*)d_in[9];
    const float* b3  = (const float*)d_in[10];
    const float* g3  = (const float*)d_in[11];
    const float* be3 = (const float*)d_in[12];
    const float* tw1 = (const float*)d_in[13];
    const float* tb1 = (const float*)d_in[14];
    const float* tg1 = (const float*)d_in[15];
    const float* tbe1= (const float*)d_in[16];
    const float* tw2 = (const float*)d_in[17];
    const float* tb2 = (const float*)d_in[18];
    const float* tg2 = (const float*)d_in[19];
    const float* tbe2= (const float*)d_in[20];
    const float* tw3 = (const float*)d_in[21];
    const float* tb3 = (const float*)d_in[22];
    const float* tg3 = (const float*)d_in[23];
    const float* tbe3= (const float*)d_in[24];
    const float* ew  = (const float*)d_in[25];
    const float* eb  = (const float*)d_in[26];
    const float* eg  = (const float*)d_in[27];
    const float* ebe = (const float*)d_in[28];
    const float* cw  = (const float*)d_in[29];
    const float* cb  = (const float*)d_in[30];

    float* out = (float*)d_out;
    float* decoded = out;                       // [16,373,64,64] = 24,444,928
    float* e2      = out + 24444928;            // [16,373]       = 5,968
    float* pred    = e2 + 5968;                 // [16,1]

    // workspace layout (floats)
    float* ws    = (float*)d_ws;
    float* bufA  = ws;                          // 48,889,856 max pre-activation
    float* bufH1 = bufA  + 48889856ull;         // 12,222,464  [16,746,32,32]
    float* bufH2 = bufH1 + 12222464ull;         //  4,583,424  [16,1119,16,16]
    float* bufE  = bufH2 + 4583424ull;          //  1,527,808  [16,1492,8,8]
    float* bufD1 = bufE  + 1527808ull;          //  4,583,424  [16,1119,16,16]
    float* bufD2 = bufD1 + 4583424ull;          // 12,222,464  [16,746,32,32]
    float* stats = bufD2 + 12222464ull;         //  4*1492
    float* s1 = stats, *s2 = stats + 1492, *mn = stats + 2984, *is = stats + 4476;

    const size_t statBytes = 2 * 1492 * sizeof(float);
    #define ZERO_STATS() (void)hipMemsetAsync(s1, 0, statBytes, stream)

    // ---- encoder stage 1: conv(373->746, 64x64) -> BN -> ReLU -> pool2 ----
    ZERO_STATS();
    gconv3x3_tiled<1, 2><<<dim3(16, G, N), TPB, 0, stream>>>(x, w1, b1, bufA, s1, s2, 64, 64, G);
    finalize_stats<<<(746 + TPB - 1) / TPB, TPB, 0, stream>>>(s1, s2, mn, is, 746, 1.0f / (N * 64 * 64));
    bn_relu_pool2<<<dim3(4, 746, N), TPB, 0, stream>>>(bufA, mn, is, g1, be1, bufH1, 746, 64, 64);

    // ---- encoder stage 2: conv(746->1119, 32x32) ----
    ZERO_STATS();
    gconv3x3_tiled<2, 3><<<dim3(4, G, N), TPB, 0, stream>>>(bufH1, w2, b2, bufA, s1, s2, 32, 32, G);
    finalize_stats<<<(1119 + TPB - 1) / TPB, TPB, 0, stream>>>(s1, s2, mn, is, 1119, 1.0f / (N * 32 * 32));
    bn_relu_pool2<<<dim3(1, 1119, N), TPB, 0, stream>>>(bufA, mn, is, g2, be2, bufH2, 1119, 32, 32);

    // ---- encoder stage 3: conv(1119->1492, 16x16) -> enc [16,1492,8,8] ----
    ZERO_STATS();
    gconv3x3_tiled<3, 4><<<dim3(1, G, N), TPB, 0, stream>>>(bufH2, w3, b3, bufA, s1, s2, 16, 16, G);
    finalize_stats<<<(1492 + TPB - 1) / TPB, TPB, 0, stream>>>(s1, s2, mn, is, 1492, 1.0f / (N * 16 * 16));
    bn_relu_pool2<<<dim3(1, 1492, N), TPB, 0, stream>>>(bufA, mn, is, g3, be3, bufE, 1492, 16, 16);

    // ---- decoder stage 1: convT(1492->1119), 8x8 -> 16x16 ----
    ZERO_STATS();
    gconvT2x2_stats<<<dim3(1, 1119, N), TPB, 0, stream>>>(bufE, tw1, tb1, bufA, s1, s2, 8, 8, 4, 3, G);
    finalize_stats<<<(1119 + TPB - 1) / TPB, TPB, 0, stream>>>(s1, s2, mn, is, 1119, 1.0f / (N * 16 * 16));
    bn_act<<<(4583424u + TPB - 1) / TPB, TPB, 0, stream>>>(bufA, mn, is, tg1, tbe1, bufD1,
                                                           (size_t)4583424u, 1119, 256, 0);

    // ---- decoder stage 2: convT(1119->746), 16x16 -> 32x32 ----
    ZERO_STATS();
    gconvT2x2_stats<<<dim3(4, 746, N), TPB, 0, stream>>>(bufD1, tw2, tb2, bufA, s1, s2, 16, 16, 3, 2, G);
    finalize_stats<<<(746 + TPB - 1) / TPB, TPB, 0, stream>>>(s1, s2, mn, is, 746, 1.0f / (N * 32 * 32));
    bn_act<<<(12222464u + TPB - 1) / TPB, TPB, 0, stream>>>(bufA, mn, is, tg2, tbe2, bufD2,
                                                            (size_t)12222464u, 746, 1024, 0);

    // ---- decoder stage 3: convT(746->373), 32x32 -> 64x64, BN+sigmoid ----
    ZERO_STATS();
    gconvT2x2_stats<<<dim3(16, 373, N), TPB, 0, stream>>>(bufD2, tw3, tb3, bufA, s1, s2, 32, 32, 2, 1, G);
    finalize_stats<<<(373 + TPB - 1) / TPB, TPB, 0, stream>>>(s1, s2, mn, is, 373, 1.0f / (N * 64 * 64));
    bn_act<<<(24444928u + TPB - 1) / TPB, TPB, 0, stream>>>(bufA, mn, is, tg3, tbe3, decoded,
                                                            (size_t)24444928u, 373, 4096, 1);

    // ---- encoder2: conv(1492->373, 8x8) -> BN -> ReLU -> pool8 -> e2 ----
    ZERO_STATS();
    gconv3x3_tiled<4, 1><<<dim3(1, G, N), TPB, 0, stream>>>(bufE, ew, eb, bufA, s1, s2, 8, 8, G);
    finalize_stats<<<(373 + TPB - 1) / TPB, TPB, 0, stream>>>(s1, s2, mn, is, 373, 1.0f / (N * 8 * 8));
    bn_relu_pool8<<<(5968 + TPB - 1) / TPB, TPB, 0, stream>>>(bufA, mn, is, eg, ebe, e2, 5968, 373);

    // ---- classifier: sigmoid(e2 @ cw^T + cb) via f32 WMMA (one full wave) --
    classifier_wmma<<<1, 32, 0, stream>>>(e2, cw, cb, pred, 373);
}